// SharedEncoder_70720931496090
// MI455X (gfx1250) — compile-verified
//
#include <hip/hip_runtime.h>
#include <hip/hip_bf16.h>

#define NN 50000
#define EE 800000
#define GG 100

typedef __attribute__((ext_vector_type(2))) float v2f;
typedef __attribute__((ext_vector_type(8))) float v8f;

__device__ __forceinline__ v8f wmma4(v2f a, v2f b, v8f c) {
    return __builtin_amdgcn_wmma_f32_16x16x4_f32(
        /*neg_a=*/false, a, /*neg_b=*/false, b,
        /*c_mod=*/(short)0, c, /*reuse_a=*/false, /*reuse_b=*/false);
}

// ---------------------------------------------------------------------------
// WMMA fp32 GEMM: C[M,Nc] = A[M,K] @ B[K,Nc] + bias, optional ReLU.
// Each wave computes a 16x64 output strip (4 x V_WMMA_F32_16X16X4_F32 tiles),
// amortizing the lane-strided A fetch over 4 WMMAs. 8 waves per 256-thread
// block cover 8 consecutive row tiles. Nc must be a multiple of 64; the
// rowTile guard is wave-uniform so EXEC is all-ones for every WMMA.
// ---------------------------------------------------------------------------
__global__ void __launch_bounds__(256)
gemm_wmma_f32(const float* __restrict__ A, const float* __restrict__ B,
              const float* __restrict__ bias, float* __restrict__ C,
              int MT, int K, int Nc, int relu)
{
    const int wave = threadIdx.x >> 5;
    const int lane = threadIdx.x & 31;
    const int rowTile = blockIdx.x * 8 + wave;
    if (rowTile >= MT) return;                  // wave-uniform exit
    const int row0  = rowTile * 16;
    const int col0  = blockIdx.y * 64;
    const int m     = lane & 15;
    const int khalf = (lane >> 4) * 2;          // 0 for lanes 0-15, 2 for 16-31
    const int n     = lane & 15;

    const float* arow = A + (size_t)(row0 + m) * K;

    v8f acc0 = {}, acc1 = {}, acc2 = {}, acc3 = {};
    for (int k = 0; k < K; k += 4) {
        v2f a;
        // A 16x4 layout: VGPR0 = K=k+khalf, VGPR1 = K=k+khalf+1 (b64 load)
        a.x = arow[k + khalf];
        a.y = arow[k + khalf + 1];
        // B 4x16 tiles: VGPR0 = row k+khalf, VGPR1 = row k+khalf+1, N striped on lanes
        const float* bp = B + (size_t)(k + khalf) * Nc + (col0 + n);
        v2f b0, b1, b2, b3;
        b0.x = bp[0];  b0.y = bp[Nc];
        b1.x = bp[16]; b1.y = bp[Nc + 16];
        b2.x = bp[32]; b2.y = bp[Nc + 32];
        b3.x = bp[48]; b3.y = bp[Nc + 48];
        acc0 = wmma4(a, b0, acc0);
        acc1 = wmma4(a, b1, acc1);
        acc2 = wmma4(a, b2, acc2);
        acc3 = wmma4(a, b3, acc3);
    }

    // C/D layout: VGPR r -> lanes 0-15: (M=r, N=lane); lanes 16-31: (M=r+8, N=lane-16)
    const int mOff = (lane >> 4) * 8;
    v8f accs[4] = {acc0, acc1, acc2, acc3};
#pragma unroll
    for (int j = 0; j < 4; ++j) {
        const int col = col0 + j * 16 + n;
        const float bc = bias ? bias[col] : 0.0f;
#pragma unroll
        for (int r = 0; r < 8; ++r) {
            float v = accs[j][r] + bc;
            if (relu) v = fmaxf(v, 0.0f);
            C[(size_t)(row0 + mOff + r) * Nc + col] = v;
        }
    }
}

// ------------------- float <-> orderable uint (for atomic max) -------------
__device__ __forceinline__ unsigned f2ord(float f) {
    unsigned u = __float_as_uint(f);
    return (u & 0x80000000u) ? ~u : (u | 0x80000000u);
}
__device__ __forceinline__ float ord2f(unsigned u) {
    return __uint_as_float((u & 0x80000000u) ? (u & 0x7fffffffu) : ~u);
}

__device__ __forceinline__ float lrelu02(float v) {
    return v >= 0.0f ? v : 0.2f * v;
}

// ---------------------------------------------------------------------------
// Pass 1: per edge, per head h: alpha[e,h] = att[h,:] . lrelu(xl[src]+xr[dst]+ea*We)
// One wave per edge; lane L handles channels [8L, 8L+8) (head = L>>3).
// Segment-max into amax via orderable-uint atomicMax (L2-resident).
// ---------------------------------------------------------------------------
__global__ void edge_alpha_kernel(const int* __restrict__ src, const int* __restrict__ dst,
                                  const float* __restrict__ ea,
                                  const float* __restrict__ We,   // [256]
                                  const float* __restrict__ att,  // [4*64]
                                  const float* __restrict__ xl, const float* __restrict__ xr,
                                  float* __restrict__ alpha, unsigned* __restrict__ amax)
{
    const int lane = threadIdx.x & 31;
    const int e = blockIdx.x * (blockDim.x >> 5) + (threadIdx.x >> 5);
    if (e >= EE) return;
    const int s = src[e], d = dst[e];
    const float w = ea[e];

    const float4* xls  = (const float4*)(xl + (size_t)s * 256);
    const float4* xrd  = (const float4*)(xr + (size_t)d * 256);
    const float4* Wev  = (const float4*)We;
    const float4* attv = (const float4*)att;

    float part = 0.0f;
#pragma unroll
    for (int q = 0; q < 2; ++q) {
        const int idx = lane * 2 + q;      // float4 index, channels idx*4..idx*4+3
        float4 a = xls[idx], b = xrd[idx], c = Wev[idx], t = attv[idx];
        part += t.x * lrelu02(a.x + b.x + w * c.x);
        part += t.y * lrelu02(a.y + b.y + w * c.y);
        part += t.z * lrelu02(a.z + b.z + w * c.z);
        part += t.w * lrelu02(a.w + b.w + w * c.w);
    }
    // reduce within each 8-lane head group (wave32)
    part += __shfl_xor(part, 1);
    part += __shfl_xor(part, 2);
    part += __shfl_xor(part, 4);
    if ((lane & 7) == 0) {
        const int h = lane >> 3;
        alpha[(size_t)e * 4 + h] = part;
        atomicMax(&amax[(size_t)d * 4 + h], f2ord(part));
    }
}

// ---------------------------------------------------------------------------
// Pass 2: ex = exp(alpha - amax[dst]); denom[dst] += ex (in-place on alpha).
// ---------------------------------------------------------------------------
__global__ void edge_exp_kernel(const int* __restrict__ dst,
                                float* __restrict__ alpha_ex,
                                const unsigned* __restrict__ amax,
                                float* __restrict__ denom)
{
    const int i = blockIdx.x * blockDim.x + threadIdx.x;   // over EE*4
    if (i >= EE * 4) return;
    const int e = i >> 2, h = i & 3;
    const int d = dst[e];
    const unsigned u = amax[(size_t)d * 4 + h];
    const float mx = (u == 0u) ? 0.0f : ord2f(u);          // untouched -> 0 (matches ref)
    const float ex = expf(alpha_ex[i] - mx);
    alpha_ex[i] = ex;
    atomicAdd(&denom[(size_t)d * 4 + h], ex);
}

// ---------------------------------------------------------------------------
// Pass 3: agg[dst,:] += xl[src,:] * (ex / (denom[dst] + 1e-16)). Wave per edge.
// ---------------------------------------------------------------------------
__global__ void edge_agg_kernel(const int* __restrict__ src, const int* __restrict__ dst,
                                const float* __restrict__ ex, const float* __restrict__ denom,
                                const float* __restrict__ xl, float* __restrict__ agg)
{
    const int lane = threadIdx.x & 31;
    const int e = blockIdx.x * (blockDim.x >> 5) + (threadIdx.x >> 5);
    if (e >= EE) return;
    const int s = src[e], d = dst[e];
    const int h = lane >> 3;
    const float a = ex[(size_t)e * 4 + h] / (denom[(size_t)d * 4 + h] + 1e-16f);
    const float* xs = xl + (size_t)s * 256 + lane * 8;
    float* od = agg + (size_t)d * 256 + lane * 8;
#pragma unroll
    for (int q = 0; q < 8; ++q)
        atomicAdd(&od[q], xs[q] * a);
}

// h[n,c] = relu(agg[n,c] + bias[c])   (in place)
__global__ void finalize_kernel(float* __restrict__ h, const float* __restrict__ bias)
{
    const int i = blockIdx.x * blockDim.x + threadIdx.x;
    if (i >= NN * 256) return;
    h[i] = fmaxf(h[i] + bias[i & 255], 0.0f);
}

// mean-pool accumulation over graphs
__global__ void pool_kernel(const float* __restrict__ h, const int* __restrict__ batch,
                            float* __restrict__ sums, float* __restrict__ cnt)
{
    const int i = blockIdx.x * blockDim.x + threadIdx.x;   // NN*256
    if (i >= NN * 256) return;
    const int n = i >> 8, c = i & 255;
    const int g = batch[n];
    atomicAdd(&sums[(size_t)g * 256 + c], h[i]);
    if (c == 0) atomicAdd(&cnt[g], 1.0f);
}

// per-graph MLP head: 256 -> 128 -> LayerNorm -> relu -> 64 -> relu
__global__ void __launch_bounds__(128)
mlp_kernel(const float* __restrict__ sums, const float* __restrict__ cnt,
           const float* __restrict__ p1W, const float* __restrict__ p1b,
           const float* __restrict__ lng, const float* __restrict__ lnb,
           const float* __restrict__ p2W, const float* __restrict__ p2b,
           float* __restrict__ out)
{
    const int g = blockIdx.x, t = threadIdx.x;   // 128 threads
    __shared__ float p[256];
    __shared__ float z[128];
    __shared__ float red[128];

    const float c = fmaxf(cnt[g], 1.0f);
    p[t]       = sums[(size_t)g * 256 + t] / c;
    p[t + 128] = sums[(size_t)g * 256 + t + 128] / c;
    __syncthreads();

    float acc = p1b[t];
    for (int k = 0; k < 256; ++k) acc += p[k] * p1W[(size_t)k * 128 + t];

    red[t] = acc; __syncthreads();
    for (int s = 64; s > 0; s >>= 1) { if (t < s) red[t] += red[t + s]; __syncthreads(); }
    const float mu = red[0] / 128.0f; __syncthreads();
    const float dv = acc - mu;
    red[t] = dv * dv; __syncthreads();
    for (int s = 64; s > 0; s >>= 1) { if (t < s) red[t] += red[t + s]; __syncthreads(); }
    const float var = red[0] / 128.0f;

    z[t] = fmaxf(dv * rsqrtf(var + 1e-5f) * lng[t] + lnb[t], 0.0f);
    __syncthreads();

    if (t < 64) {
        float o = p2b[t];
        for (int k = 0; k < 128; ++k) o += z[k] * p2W[(size_t)k * 64 + t];
        out[(size_t)g * 64 + t] = fmaxf(o, 0.0f);
    }
}

// ---------------------------------------------------------------------------
extern "C" void kernel_launch(void* const* d_in, const int* in_sizes, int n_in,
                              void* d_out, int out_size, void* d_ws, size_t ws_size,
                              hipStream_t stream)
{
    const float* x    = (const float*)d_in[0];
    const int*   eidx = (const int*)d_in[1];
    const int*   src  = eidx;
    const int*   dst  = eidx + EE;
    const float* ea   = (const float*)d_in[2];
    const int*   batch= (const int*)d_in[3];
    const float* encW = (const float*)d_in[4];
    const float* encB = (const float*)d_in[5];
    const float* g1Wl = (const float*)d_in[6];
    const float* g1bl = (const float*)d_in[7];
    const float* g1Wr = (const float*)d_in[8];
    const float* g1br = (const float*)d_in[9];
    const float* g1We = (const float*)d_in[10];
    const float* g1att= (const float*)d_in[11];
    const float* g1bias=(const float*)d_in[12];
    const float* g2Wl = (const float*)d_in[13];
    const float* g2bl = (const float*)d_in[14];
    const float* g2Wr = (const float*)d_in[15];
    const float* g2br = (const float*)d_in[16];
    const float* g2We = (const float*)d_in[17];
    const float* g2att= (const float*)d_in[18];
    const float* g2bias=(const float*)d_in[19];
    const float* p1W  = (const float*)d_in[20];
    const float* p1b  = (const float*)d_in[21];
    const float* lng  = (const float*)d_in[22];
    const float* lnb  = (const float*)d_in[23];
    const float* p2W  = (const float*)d_in[24];
    const float* p2b  = (const float*)d_in[25];
    float* out = (float*)d_out;

    // workspace layout (floats)
    float* ws = (float*)d_ws;
    size_t o = 0;
    float*    h0    = ws + o; o += (size_t)NN * 64;
    float*    xl    = ws + o; o += (size_t)NN * 256;
    float*    xr    = ws + o; o += (size_t)NN * 256;
    float*    alpha = ws + o; o += (size_t)EE * 4;    // reused as ex
    unsigned* amax  = (unsigned*)(ws + o); o += (size_t)NN * 4;
    float*    denom = ws + o; o += (size_t)NN * 4;
    float*    agg   = ws + o; o += (size_t)NN * 256;  // reused as h1 / agg2 / h2
    float*    psum  = ws + o; o += (size_t)GG * 256;
    float*    pcnt  = ws + o; o += 128;

    const int MT = NN / 16;                  // 3125 row tiles (exact)
    const int gemmBlks = (MT + 7) / 8;       // 8 waves (row tiles) per block
    const int wavesPerBlk = 8;               // 256 threads = 8 waves
    const int edgeBlks = (EE + wavesPerBlk - 1) / wavesPerBlk;

    // ---- encoder: h0 = relu(x @ encW + encB) ----
    gemm_wmma_f32<<<dim3(gemmBlks, 1), 256, 0, stream>>>(x, encW, encB, h0, MT, 8, 64, 1);

    // ================= GATv2 layer 1 (in=64) =================
    gemm_wmma_f32<<<dim3(gemmBlks, 4), 256, 0, stream>>>(h0, g1Wl, g1bl, xl, MT, 64, 256, 0);
    gemm_wmma_f32<<<dim3(gemmBlks, 4), 256, 0, stream>>>(h0, g1Wr, g1br, xr, MT, 64, 256, 0);

    hipMemsetAsync(amax, 0, (size_t)NN * 4 * sizeof(unsigned), stream);
    hipMemsetAsync(denom, 0, (size_t)NN * 4 * sizeof(float), stream);
    hipMemsetAsync(agg, 0, (size_t)NN * 256 * sizeof(float), stream);

    edge_alpha_kernel<<<edgeBlks, 256, 0, stream>>>(src, dst, ea, g1We, g1att, xl, xr, alpha, amax);
    edge_exp_kernel<<<(EE * 4 + 255) / 256, 256, 0, stream>>>(dst, alpha, amax, denom);
    edge_agg_kernel<<<edgeBlks, 256, 0, stream>>>(src, dst, alpha, denom, xl, agg);
    finalize_kernel<<<(NN * 256 + 255) / 256, 256, 0, stream>>>(agg, g1bias);   // agg -> h1

    // ================= GATv2 layer 2 (in=256) =================
    gemm_wmma_f32<<<dim3(gemmBlks, 4), 256, 0, stream>>>(agg, g2Wl, g2bl, xl, MT, 256, 256, 0);
    gemm_wmma_f32<<<dim3(gemmBlks, 4), 256, 0, stream>>>(agg, g2Wr, g2br, xr, MT, 256, 256, 0);

    hipMemsetAsync(amax, 0, (size_t)NN * 4 * sizeof(unsigned), stream);
    hipMemsetAsync(denom, 0, (size_t)NN * 4 * sizeof(float), stream);
    hipMemsetAsync(agg, 0, (size_t)NN * 256 * sizeof(float), stream);           // h1 consumed

    edge_alpha_kernel<<<edgeBlks, 256, 0, stream>>>(src, dst, ea, g2We, g2att, xl, xr, alpha, amax);
    edge_exp_kernel<<<(EE * 4 + 255) / 256, 256, 0, stream>>>(dst, alpha, amax, denom);
    edge_agg_kernel<<<edgeBlks, 256, 0, stream>>>(src, dst, alpha, denom, xl, agg);
    finalize_kernel<<<(NN * 256 + 255) / 256, 256, 0, stream>>>(agg, g2bias);   // agg -> h2

    // ================= pool + MLP head =================
    hipMemsetAsync(psum, 0, (size_t)GG * 256 * sizeof(float), stream);
    hipMemsetAsync(pcnt, 0, 128 * sizeof(float), stream);
    pool_kernel<<<(NN * 256 + 255) / 256, 256, 0, stream>>>(agg, batch, psum, pcnt);
    mlp_kernel<<<GG, 128, 0, stream>>>(psum, pcnt, p1W, p1b, lng, lnb, p2W, p2b, out);
}